// SharedRadialLinearTransform_42571715838069
// MI455X (gfx1250) — compile-verified
//
#include <hip/hip_runtime.h>

// CDNA5 / gfx1250, wave32.
// Compute:  V_WMMA_F32_16X16X4_F32 (3 chained per 16-wide m-tile, K=12 exact).
// Input:    Tensor Data Mover (tensor_load_to_lds) pulls each wave's strided
//           12x128 f32 tile into LDS in ONE instruction (replaces 48 b32 loads),
//           with TDM row padding (+8 dwords) for bank-conflict-free ds reads.
// Output:   direct coalesced global_store_b32 (rows are non-contiguous in VGPRs).

typedef float        v2f __attribute__((ext_vector_type(2)));
typedef float        v8f __attribute__((ext_vector_type(8)));
typedef unsigned int v4u __attribute__((ext_vector_type(4)));
typedef int          v8i __attribute__((ext_vector_type(8)));
typedef int          v4i __attribute__((ext_vector_type(4)));

#define N_NODES 2048
#define RDIM    12
#define ADIM    20
#define EDIM    128
#define R_STRIDE (ADIM * EDIM)        // 2560 floats between radial slices (global)
#define N_STRIDE (RDIM * ADIM * EDIM) // 30720 floats between nodes (global)
#define LROW     136                  // LDS row stride in floats: 128 + 8 (TDM pad)
#define WPB      8                    // waves per block

__device__ __forceinline__ int group_of(int a) {
    return (a >= 1) + (a >= 4) + (a >= 10);   // group sizes [1,3,6,10]
}

__global__ __launch_bounds__(256)
void SharedRadialLinearTransform_wmma(const float* __restrict__ x,
                                      const float* __restrict__ W,
                                      float* __restrict__ out) {
    // per-wave LDS staging tile: 12 padded rows of 136 floats (6528 B); 8 waves
    __shared__ float xs[WPB][RDIM * LROW];    // 52,224 B total

    const int tid  = blockIdx.x * blockDim.x + threadIdx.x;
    const int wave = tid >> 5;                // one wave per (node, angular) pair
    const int lane = tid & 31;
    const int half = lane >> 4;
    const int col  = lane & 15;
    const int wib  = (threadIdx.x >> 5);      // wave index in block

    const int n = wave / ADIM;
    const int a = wave % ADIM;
    const int g = group_of(a);

    // ================= TDM: load 12x128 f32 tile -> LDS =================
    // D# group 0: count=1 | lds_addr | global_addr(57b) | type=2
    const unsigned long long ga =
        (unsigned long long)(uintptr_t)(x + (size_t)n * N_STRIDE + (size_t)a * EDIM);
    const unsigned int lds_base = (unsigned int)(uintptr_t)(&xs[wib][0]);

    const unsigned int g0w0 = 1u;                                   // count=1
    const unsigned int g0w1 = (unsigned int)__builtin_amdgcn_readfirstlane((int)lds_base);
    const unsigned int g0w2 = (unsigned int)__builtin_amdgcn_readfirstlane((int)(unsigned int)ga);
    const unsigned int g0w3 = ((unsigned int)__builtin_amdgcn_readfirstlane((int)(unsigned int)(ga >> 32))
                               & 0x01FFFFFFu) | (2u << 30);         // addr[56:32] | type=2
    v4u g0 = {g0w0, g0w1, g0w2, g0w3};

    // D# group 1:
    //  d0: wg_mask=0 | data_size=2(4B)<<16 | pad_en<<20 | pad_interval=6(128dw)<<22
    //      | pad_amount=7(8dw)<<25
    //  d1: atomic_barrier_addr=0 | tensor_dim0.lo16=128 <<16
    //  d2: tensor_dim0.hi16=0 | tensor_dim1.lo16=12 <<16
    //  d3: tensor_dim1.hi16=0 | tile_dim0=128 <<16
    //  d4: tile_dim1=12 | tile_dim2=0
    //  d5: tensor_dim0_stride.lo32 = 2560
    //  d6: tensor_dim0_stride.hi16=0 | tensor_dim1_stride.lo16 <<16
    //  d7: tensor_dim1_stride.hi32
    v8i g1 = {(int)((2u << 16) | (1u << 20) | (6u << 22) | (7u << 25)),
              (int)(128u << 16),
              (int)(12u << 16),
              (int)(128u << 16),
              (int)12u,
              (int)R_STRIDE,
              (int)((unsigned int)N_STRIDE << 16),
              (int)((unsigned int)N_STRIDE >> 16)};
    v4i g2 = {0, 0, 0, 0};   // 2-D tile: tile_dim2/3 = 0 (unused)
    v4i g3 = {0, 0, 0, 0};
    v8i g4 = {0, 0, 0, 0, 0, 0, 0, 0};   // extension slot (clang-23 6-arg form)

    // amdgpu-toolchain (clang-23) 6-arg form:
    // (uint32x4, int32x8, int32x4, int32x4, int32x8, i32 cpol)
    __builtin_amdgcn_tensor_load_to_lds(g0, g1, g2, g3, g4, /*cpol*/0);

    // ---- A fragment (16x4 f32): A[s][r] = W[g][r][s]; built while TDM runs ----
    const float* __restrict__ Wg = W + g * (RDIM * RDIM);
    const int  cs    = (col < RDIM) ? col : 0;
    const bool valid = (col < RDIM);
    v2f afrag[3];
    #pragma unroll
    for (int c = 0; c < 3; ++c) {
        #pragma unroll
        for (int v = 0; v < 2; ++v) {
            const int r = 4 * c + v + 2 * half;      // r in [0,12)
            const float wv = Wg[r * RDIM + cs];
            afrag[c][v] = valid ? wv : 0.0f;         // v_cndmask, no exec masking
        }
    }

    // output base with lane-half folded in: all store offsets are immediates
    float* __restrict__ ob =
        out + (size_t)n * N_STRIDE + a * EDIM + col + (8 * half) * R_STRIDE;

    // LDS read base with lane-half folded in: row r at float index r*LROW
    const float* __restrict__ xsw = &xs[wib][0] + col + (2 * half) * LROW;

    __builtin_amdgcn_s_wait_tensorcnt(0);    // tile resident in LDS

    // ================= 8 m-tiles of 16 =================
    #pragma unroll
    for (int t = 0; t < 8; ++t) {
        const int mofs = t * 16;

        v8f acc = {};
        #pragma unroll
        for (int c = 0; c < 3; ++c) {
            v2f bfrag;                                // B: K = v + 2*half, N = col
            #pragma unroll
            for (int v = 0; v < 2; ++v) {
                bfrag[v] = xsw[(4 * c + v) * LROW + mofs];   // ds load, imm offset
            }
            acc = __builtin_amdgcn_wmma_f32_16x16x4_f32(
                false, afrag[c], false, bfrag, (short)0, acc, false, false);
        }

        // rows d=0..3 valid for all lanes (s = d or d+8 < 12)
        #pragma unroll
        for (int d = 0; d < 4; ++d) {
            ob[d * R_STRIDE + mofs] = acc[d];
        }
        // rows d=4..7 valid only for lanes 0-15 (one toggle, clause of 4)
        if (half == 0) {
            #pragma unroll
            for (int d = 4; d < 8; ++d) {
                ob[d * R_STRIDE + mofs] = acc[d];
            }
        }
    }
}

extern "C" void kernel_launch(void* const* d_in, const int* in_sizes, int n_in,
                              void* d_out, int out_size, void* d_ws, size_t ws_size,
                              hipStream_t stream) {
    const float* x = (const float*)d_in[0];
    const float* W = (const float*)d_in[1];
    float* out     = (float*)d_out;

    const int total_waves = N_NODES * ADIM;           // 40960
    const int threads     = 256;                      // 8 waves/block
    const int blocks      = total_waves / (threads / 32); // 5120, exact

    SharedRadialLinearTransform_wmma<<<blocks, threads, 0, stream>>>(x, W, out);
}